// InvDirectImageAlign_42863773614751
// MI455X (gfx1250) — compile-verified
//
#include <hip/hip_runtime.h>
#include <hip/hip_bf16.h>

#define HD 320
#define WD 448
#define HW (HD*WD)                 // 143360 = 4480 * 32
#define NB 16
#define CHUNKS_PER_BATCH (HW/32)   // 4480
#define WAVES_PER_BATCH 448
#define WAVES_PER_BLOCK 4
#define BLOCKS_PER_BATCH (WAVES_PER_BATCH/WAVES_PER_BLOCK)  // 112
#define TOTAL_BLOCKS (NB*BLOCKS_PER_BATCH)                  // 1792
#define GN_ITERS 5
#define LAM 0.01f
#define HDELTA 0.1f
#define FEPS 1e-6f

typedef float v2f __attribute__((ext_vector_type(2)));
typedef float v4f __attribute__((ext_vector_type(4)));
typedef float v8f __attribute__((ext_vector_type(8)));

__device__ __forceinline__ int imin(int a, int b) { return a < b ? a : b; }
__device__ __forceinline__ int imax(int a, int b) { return a > b ? a : b; }

// ---------------------------------------------------------------------------
// se3_exp (matches the reference, including small-angle branches)
// ---------------------------------------------------------------------------
__device__ __forceinline__ void se3exp_dev(const float* v, const float* w, float* T) {
  float th2  = w[0]*w[0] + w[1]*w[1] + w[2]*w[2];
  float th2c = fmaxf(th2, 1e-16f);
  float th   = sqrtf(th2c);
  bool  small_ = th2 < 1e-10f;
  float A  = small_ ? (1.f  - th2*(1.f/6.f))   : (sinf(th)/th);
  float Bc = small_ ? (0.5f - th2*(1.f/24.f))  : ((1.f - cosf(th))/th2c);
  float Cc = small_ ? ((1.f/6.f) - th2*(1.f/120.f)) : ((1.f - A)/th2c);
  float K[9] = {0.f, -w[2], w[1],   w[2], 0.f, -w[0],   -w[1], w[0], 0.f};
  float K2[9];
#pragma unroll
  for (int i = 0; i < 3; i++)
#pragma unroll
    for (int j = 0; j < 3; j++) {
      float s = 0.f;
#pragma unroll
      for (int k = 0; k < 3; k++) s += K[i*3+k]*K[k*3+j];
      K2[i*3+j] = s;
    }
  float R[9], V[9];
#pragma unroll
  for (int i = 0; i < 3; i++)
#pragma unroll
    for (int j = 0; j < 3; j++) {
      float id = (i == j) ? 1.f : 0.f;
      R[i*3+j] = id + A *K[i*3+j] + Bc*K2[i*3+j];
      V[i*3+j] = id + Bc*K[i*3+j] + Cc*K2[i*3+j];
    }
#pragma unroll
  for (int i = 0; i < 3; i++) {
    T[i*4+0] = R[i*3+0]; T[i*4+1] = R[i*3+1]; T[i*4+2] = R[i*3+2];
    T[i*4+3] = V[i*3+0]*v[0] + V[i*3+1]*v[1] + V[i*3+2]*v[2];
  }
  T[12] = 0.f; T[13] = 0.f; T[14] = 0.f; T[15] = 1.f;
}

// ---------------------------------------------------------------------------
// init: T_init = T_cur = se3_exp(pose_twist); zero 7x7 accumulators
// ---------------------------------------------------------------------------
__global__ void init_kernel(const float* __restrict__ xi,
                            float* __restrict__ Tinit, float* __restrict__ Tcur,
                            float* __restrict__ G) {
  int b = threadIdx.x;
  if (b >= NB) return;
  float vv[3] = {xi[b*6+0], xi[b*6+1], xi[b*6+2]};
  float ww[3] = {xi[b*6+3], xi[b*6+4], xi[b*6+5]};
  float T[16];
  se3exp_dev(vv, ww, T);
#pragma unroll
  for (int i = 0; i < 16; i++) { Tinit[b*16+i] = T[i]; Tcur[b*16+i] = T[i]; }
  for (int i = 0; i < 49; i++) G[b*49+i] = 0.f;
}

// ---------------------------------------------------------------------------
// Main GN accumulation kernel.
// Per lane: one pixel -> 4 weighted residual rows (K=4). Staged to LDS in
// [m][k] order, then one V_WMMA_F32_16X16X4_F32 per pixel accumulates the
// 7x7 (JtWJ | JtWr) block into the wave's f32 accumulator.
// ---------------------------------------------------------------------------
__global__ __launch_bounds__(128)
void gn_accum_kernel(const float* __restrict__ I0, const float* __restrict__ I1,
                     const float* __restrict__ invD0, const float* __restrict__ invD1,
                     const float* __restrict__ intr,
                     const float* __restrict__ Tinit, const float* __restrict__ Tcur,
                     float* __restrict__ G) {
  __shared__ float smem[WAVES_PER_BLOCK * 32 * 64];   // 32 KB

  const int lane = threadIdx.x & 31;
  const int wib  = threadIdx.x >> 5;
  const int b    = blockIdx.x / BLOCKS_PER_BATCH;
  const int wb   = (blockIdx.x % BLOCKS_PER_BATCH) * WAVES_PER_BLOCK + wib;

  const float fx = intr[b*4+0], fy = intr[b*4+1];
  const float cx = intr[b*4+2], cy = intr[b*4+3];
  const float* Tc = Tcur  + b*16;
  const float* Tj = Tinit + b*16;
  float Rc[12], Rj[12];
#pragma unroll
  for (int i = 0; i < 12; i++) { Rc[i] = Tc[i]; Rj[i] = Tj[i]; }

  const float* I0b = I0 + (size_t)b*3*HW;
  const float* I1b = I1 + (size_t)b*3*HW;
  const float* D0b = invD0 + (size_t)b*HW;
  const float* D1b = invD1 + (size_t)b*HW;

  float* wlds = smem + wib*2048;
  const int   r16    = lane & 15;
  const float opmask = (r16 < 8) ? 1.0f : 0.0f;      // rows/cols 8..15 -> 0
  const int   ldoff  = (r16 & 7)*4 + ((lane >> 4) << 1);  // [m][k] pair (k=2h,2h+1)

  v8f acc = {0.f,0.f,0.f,0.f,0.f,0.f,0.f,0.f};

  for (int chunk = wb; chunk < CHUNKS_PER_BATCH; chunk += WAVES_PER_BATCH) {
    const int pix = chunk*32 + lane;
    const int py  = pix / WD;
    const int px  = pix - py*WD;

    // ---- geometry (current pose for warp, initial pose for Jacobians) ----
    const float iD1raw = D1b[pix];
    const float z1 = 1.0f / fmaxf(iD1raw, FEPS);
    const float x1 = (((float)px) - cx)/fx * z1;
    const float y1 = (((float)py) - cy)/fy * z1;

    float X0 = Rc[0]*x1 + Rc[1]*y1 + Rc[2]*z1 + Rc[3];
    float Y0 = Rc[4]*x1 + Rc[5]*y1 + Rc[6]*z1 + Rc[7];
    float Z0 = Rc[8]*x1 + Rc[9]*y1 + Rc[10]*z1 + Rc[11];
    float z0s = (fabsf(Z0) > FEPS) ? Z0 : FEPS;
    float iz  = 1.0f / z0s;
    float u0  = fx*X0*iz + cx;
    float v0  = fy*Y0*iz + cy;
    bool  vmb = (Z0 > FEPS) && (iD1raw > FEPS) &&
                (u0 > 0.f) && (u0 < (float)(WD-1)) &&
                (v0 > 0.f) && (v0 < (float)(HD-1));

    float Xj = Rj[0]*x1 + Rj[1]*y1 + Rj[2]*z1 + Rj[3];
    float Yj = Rj[4]*x1 + Rj[5]*y1 + Rj[6]*z1 + Rj[7];
    float Zj = Rj[8]*x1 + Rj[9]*y1 + Rj[10]*z1 + Rj[11];
    float zjs = (fabsf(Zj) > FEPS) ? Zj : FEPS;
    float izj = 1.0f / zjs;
    float aj = fx*izj, bj = fy*izj;
    float cj = -fx*Xj*izj*izj, dj = -fy*Yj*izj*izj;

    // Jw = Jp @ [I3 | -skew(X0j)], Jt_z = row 2 of Jt
    float Jw0[6], Jw1[6], Jtz[6];
    Jw0[0]=aj;  Jw0[1]=0.f; Jw0[2]=cj;
    Jw0[3]=cj*Yj;          Jw0[4]=aj*Zj - cj*Xj;  Jw0[5]=-aj*Yj;
    Jw1[0]=0.f; Jw1[1]=bj;  Jw1[2]=dj;
    Jw1[3]=-bj*Zj + dj*Yj; Jw1[4]=-dj*Xj;         Jw1[5]=bj*Xj;
    Jtz[0]=0.f; Jtz[1]=0.f; Jtz[2]=1.f; Jtz[3]=Yj; Jtz[4]=-Xj; Jtz[5]=0.f;

    // ---- bilinear sample of I0, invD0 and their on-the-fly gradients ----
    float gx = fminf(fmaxf(2.f*(u0/(float)(WD-1)) - 1.f, -2.f), 2.f);
    float gy = fminf(fmaxf(2.f*(v0/(float)(HD-1)) - 1.f, -2.f), 2.f);
    float ixc = (gx + 1.f)*0.5f*(float)(WD-1);
    float iyc = (gy + 1.f)*0.5f*(float)(HD-1);
    float x0f = floorf(ixc), y0f = floorf(iyc);
    float wx1 = ixc - x0f,   wy1 = iyc - y0f;

    float sI[3] = {0,0,0}, sdx[3] = {0,0,0}, sdy[3] = {0,0,0};
    float sD = 0.f, sDx = 0.f, sDy = 0.f;
#pragma unroll
    for (int corner = 0; corner < 4; ++corner) {
      float xf = x0f + (float)(corner & 1);
      float yf = y0f + (float)(corner >> 1);
      float cwgt = ((corner & 1) ? wx1 : (1.f - wx1)) *
                   ((corner >> 1) ? wy1 : (1.f - wy1));
      float m = (xf >= 0.f && xf <= (float)(WD-1) &&
                 yf >= 0.f && yf <= (float)(HD-1)) ? cwgt : 0.f;
      int xc = imin(imax((int)xf, 0), WD-1);
      int yc = imin(imax((int)yf, 0), HD-1);
      int xp = imin(xc+1, WD-1), xm = imax(xc-1, 0);
      int yp = imin(yc+1, HD-1), ym = imax(yc-1, 0);
      int rowc = yc*WD;
#pragma unroll
      for (int c = 0; c < 3; c++) {
        const float* p = I0b + c*HW;
        float vc  = p[rowc + xc];
        float dxv = 0.5f*(p[rowc + xp] - p[rowc + xm]);
        float dyv = 0.5f*(p[yp*WD + xc] - p[ym*WD + xc]);
        sI[c] += m*vc; sdx[c] += m*dxv; sdy[c] += m*dyv;
      }
      float vD  = D0b[rowc + xc];
      float dDx = 0.5f*(D0b[rowc + xp] - D0b[rowc + xm]);
      float dDy = 0.5f*(D0b[yp*WD + xc] - D0b[ym*WD + xc]);
      sD += m*vD; sDx += m*dDx; sDy += m*dDy;
    }

    // ---- residual rows and Jacobian rows (3 RGB + lambda*depth) ----
    float rr[4], wts[4], jr[4][6];
#pragma unroll
    for (int c = 0; c < 3; c++) {
      rr[c] = vmb ? (I1b[c*HW + pix] - sI[c]) : 1e-6f;
#pragma unroll
      for (int n = 0; n < 6; n++) jr[c][n] = -(sdx[c]*Jw0[n] + sdy[c]*Jw1[n]);
    }
    rr[3] = LAM * (vmb ? (iz - sD) : 1e-6f);
#pragma unroll
    for (int n = 0; n < 6; n++) jr[3][n] = -LAM*((sDx*Jw0[n] + sDy*Jw1[n]) - Jtz[n]);
#pragma unroll
    for (int k = 0; k < 4; k++)
      wts[k] = fminf(1.0f, HDELTA / fmaxf(fabsf(rr[k]), FEPS));

    // ---- stage to LDS: WJ[m][k] (A operand) then JJ[m][k] (B operand) ----
    float* pb = wlds + lane*64;
#pragma unroll
    for (int mrow = 0; mrow < 8; ++mrow) {
      v4f wjv, jv;
#pragma unroll
      for (int k = 0; k < 4; k++) {
        float jval = (mrow < 6) ? jr[k][mrow] : ((mrow == 6) ? rr[k] : 0.f);
        jv[k]  = jval;
        wjv[k] = wts[k]*jval;
      }
      *(v4f*)(pb + mrow*4)      = wjv;
      *(v4f*)(pb + 32 + mrow*4) = jv;
    }

    asm volatile("" ::: "memory");  // keep DS stores before DS loads (same-wave in-order)

    // ---- one WMMA per pixel: acc[m][n] += sum_k w_k j_k[m] * j_k[n] ----
    for (int g = 0; g < 32; ++g) {
      const float* gb = wlds + g*64;
      v2f av = *(const v2f*)(gb + ldoff);
      v2f bv = *(const v2f*)(gb + 32 + ldoff);
      av[0] *= opmask; av[1] *= opmask;
      bv[0] *= opmask; bv[1] *= opmask;
      acc = __builtin_amdgcn_wmma_f32_16x16x4_f32(
          /*neg_a=*/false, av, /*neg_b=*/false, bv,
          /*c_mod=*/(short)0, acc, /*reuse_a=*/false, /*reuse_b=*/false);
    }
  }

  // C layout: VGPR q = row M=q for lanes 0..15 (N = lane). Need M,N in 0..6.
  if (lane < 7) {
    float* Gb = G + b*49;
#pragma unroll
    for (int q = 0; q < 7; q++) atomicAdd(&Gb[q*7 + lane], acc[q]);
  }
}

// ---------------------------------------------------------------------------
// Per-batch 6x6 solve (Gauss-Jordan w/ partial pivot), pose update, re-zero G
// ---------------------------------------------------------------------------
__global__ void solve_kernel(float* __restrict__ Tcur, float* __restrict__ G,
                             float* __restrict__ out, int last) {
  int b = threadIdx.x;
  if (b >= NB) return;
  float* Gb = G + b*49;
  float M[6][7];
  float tr = 0.f;
#pragma unroll
  for (int i = 0; i < 6; i++) {
#pragma unroll
    for (int j = 0; j < 6; j++) M[i][j] = Gb[i*7+j];
    M[i][6] = Gb[i*7+6];
    tr += Gb[i*7+i];
  }
#pragma unroll
  for (int i = 0; i < 6; i++) M[i][i] += tr*1e-6f;

  for (int col = 0; col < 6; ++col) {
    int piv = col; float best = fabsf(M[col][col]);
    for (int r = col+1; r < 6; r++) {
      float vv = fabsf(M[r][col]);
      if (vv > best) { best = vv; piv = r; }
    }
    if (piv != col)
      for (int j = 0; j < 7; j++) { float t = M[col][j]; M[col][j] = M[piv][j]; M[piv][j] = t; }
    float inv = 1.0f / M[col][col];
    for (int j = 0; j < 7; j++) M[col][j] *= inv;
    for (int r = 0; r < 6; r++) {
      if (r == col) continue;
      float f = M[r][col];
      for (int j = 0; j < 7; j++) M[r][j] -= f*M[col][j];
    }
  }

  float nv[3] = {-M[0][6], -M[1][6], -M[2][6]};
  float nw[3] = {-M[3][6], -M[4][6], -M[5][6]};
  float E[16];
  se3exp_dev(nv, nw, E);

  float Told[16], Tn[16];
#pragma unroll
  for (int i = 0; i < 16; i++) Told[i] = Tcur[b*16+i];
#pragma unroll
  for (int i = 0; i < 4; i++)
#pragma unroll
    for (int j = 0; j < 4; j++) {
      float s = 0.f;
#pragma unroll
      for (int k = 0; k < 4; k++) s += Told[i*4+k]*E[k*4+j];
      Tn[i*4+j] = s;
    }
#pragma unroll
  for (int i = 0; i < 16; i++) Tcur[b*16+i] = Tn[i];
  for (int i = 0; i < 49; i++) Gb[i] = 0.f;
  if (last) {
#pragma unroll
    for (int i = 0; i < 16; i++) out[b*16+i] = Tn[i];
  }
}

// ---------------------------------------------------------------------------
extern "C" void kernel_launch(void* const* d_in, const int* in_sizes, int n_in,
                              void* d_out, int out_size, void* d_ws, size_t ws_size,
                              hipStream_t stream) {
  const float* pose  = (const float*)d_in[0];
  const float* I0    = (const float*)d_in[1];
  const float* I1    = (const float*)d_in[2];
  const float* invD0 = (const float*)d_in[3];
  const float* invD1 = (const float*)d_in[4];
  const float* intr  = (const float*)d_in[5];
  float* ws    = (float*)d_ws;
  float* Tinit = ws;          // 256 floats
  float* Tcur  = ws + 256;    // 256 floats
  float* G     = ws + 512;    // 16 * 49 floats

  init_kernel<<<1, NB, 0, stream>>>(pose, Tinit, Tcur, G);
  for (int it = 0; it < GN_ITERS; ++it) {
    gn_accum_kernel<<<TOTAL_BLOCKS, 128, 0, stream>>>(
        I0, I1, invD0, invD1, intr, Tinit, Tcur, G);
    solve_kernel<<<1, NB, 0, stream>>>(Tcur, G, (float*)d_out,
                                       (it == GN_ITERS-1) ? 1 : 0);
  }
  (void)in_sizes; (void)n_in; (void)out_size; (void)ws_size;
}